// MonoFlexProcessor_14113262535332
// MI455X (gfx1250) — compile-verified
//
#include <hip/hip_runtime.h>
#include <stdint.h>

// ---------------------------------------------------------------------------
// MonoFlex post-processing for MI455X (gfx1250).
//   B=16, NCLS=3, CREG=50, H=96, W=320, K=100, DOWN=4, THRESH=0.2
// Pipeline: init counters -> TDM-staged NMS+threshold+append ->
//           per-batch iterated-max top-100 -> sparse gather + decode.
// Output: (B*K, 13) = [cls, score, box2d(4), dims(3), loc3d(3), depth_err]
// ---------------------------------------------------------------------------

#define B_     16
#define NCLS_  3
#define CREG_  50
#define H_     96
#define W_     320
#define K_     100
#define DOWN_  4
#define NCOL_  13
#define IMG_W_ (W_ * DOWN_)
#define IMG_H_ (H_ * DOWN_)
#define CAND_CAP 16384
#define ROWS_PER_BLK 24
#define LOGIT_TH (-1.3862943611f)   /* log(0.2/0.8): sigmoid(x)>=0.2 <=> x>=this */

typedef uint32_t u32x4 __attribute__((ext_vector_type(4)));
typedef uint32_t u32x8 __attribute__((ext_vector_type(8)));

// ---------------------------------------------------------------------------
__global__ void init_kernel(unsigned int* __restrict__ cnt) {
  if (threadIdx.x < B_) cnt[threadIdx.x] = 0u;
}

// ---------------------------------------------------------------------------
// One workgroup per (batch, class, row-block). Stages a (rows+halo) x 320 f32
// tile into LDS via the Tensor Data Mover, then does sigmoid-free NMS on
// logits and appends thresholded candidates with one atomic per wave.
// Candidate key: [63:32]=float bits of sigmoid score (positive -> orderable),
//                [31:0]=~ (c*H*W + y*W + x)  -> descending u64 sort gives
//                score-desc, then (class, spatial index)-asc tie order.
// ---------------------------------------------------------------------------
__global__ __launch_bounds__(256)
void nms_topcand_kernel(const float* __restrict__ hmp,
                        unsigned long long* __restrict__ cand,
                        unsigned int* __restrict__ cnt) {
  const int blk   = blockIdx.x;
  const int rb    = blk & 3;                 // 4 row-blocks of 24 rows
  const int plane = blk >> 2;                // b*NCLS + c
  const int b     = plane / NCLS_;
  const int c     = plane - b * NCLS_;
  const int r0    = rb * ROWS_PER_BLK;
  const int r_lo  = (r0 - 1 < 0) ? 0 : (r0 - 1);
  const int r_hi  = (r0 + ROWS_PER_BLK > H_ - 1) ? (H_ - 1) : (r0 + ROWS_PER_BLK);
  const int nrows = r_hi - r_lo + 1;         // <= 26

  __shared__ float sh[ROWS_PER_BLK + 2][W_]; // 26*320*4 = 33280 B

  // ---- TDM: one DMA for the whole tile; wave 0 issues (TDM ignores EXEC) --
  if (threadIdx.x < 32) {
    const uint64_t ga  = (uint64_t)(uintptr_t)(hmp + ((size_t)plane * H_ + r_lo) * W_);
    const uint32_t lds = (uint32_t)(uintptr_t)(&sh[0][0]); // low 32b of generic LDS addr = LDS offset

    u32x4 g0; u32x8 g1; u32x4 g2; u32x4 g3;
    // D# group0: count=1 | lds_addr | global_addr[56:0] | type=2
    g0[0] = 1u;
    g0[1] = lds;
    g0[2] = (uint32_t)ga;
    g0[3] = (uint32_t)((ga >> 32) & 0x01FFFFFFu) | (2u << 30);
    // D# group1: data_size=4B, tensor 320 x nrows, tile 320 x nrows, stride 320
    g1[0] = 2u << 16;                                                  // data_size=2 (4B)
    g1[1] = ((uint32_t)W_ & 0xFFFFu) << 16;                            // tensor_dim0 lo16
    g1[2] = (((uint32_t)W_ >> 16) & 0xFFFFu)
          | (((uint32_t)nrows & 0xFFFFu) << 16);                       // dim0 hi | dim1 lo
    g1[3] = (((uint32_t)nrows >> 16) & 0xFFFFu)
          | (((uint32_t)W_ & 0xFFFFu) << 16);                          // dim1 hi | tile_dim0
    g1[4] = (uint32_t)nrows & 0xFFFFu;                                 // tile_dim1 | tile_dim2=0
    g1[5] = (uint32_t)W_;                                              // tensor_dim0_stride lo32
    g1[6] = 0u;
    g1[7] = 0u;
    g2[0] = 0u; g2[1] = 0u; g2[2] = 0u; g2[3] = 0u;                    // dims 3..5 unused
    g3[0] = 0u; g3[1] = 0u; g3[2] = 0u; g3[3] = 0u;

    asm volatile("tensor_load_to_lds %0, %1, %2, %3"
                 :: "s"(g0), "s"(g1), "s"(g2), "s"(g3)
                 : "memory");
    __builtin_amdgcn_s_wait_tensorcnt(0);
  }
  __syncthreads();

  // ---- NMS + threshold + wave-aggregated append ---------------------------
  // Uniform trip count (7680/256 == 30) -> ballot participation is uniform.
  for (int p = threadIdx.x; p < ROWS_PER_BLK * W_; p += 256) {
    const int rr = p / W_;
    const int x  = p - rr * W_;
    const int r  = r0 + rr;
    const int ls = r - r_lo;
    const float v = sh[ls][x];

    bool want = (v >= LOGIT_TH);             // sigmoid(v) >= 0.2
    if (want) {
      #pragma unroll
      for (int dy = -1; dy <= 1; ++dy) {
        const int y2 = r + dy;
        if (y2 < 0 || y2 >= H_) continue;    // 'SAME' pad = -inf
        #pragma unroll
        for (int dx = -1; dx <= 1; ++dx) {
          if (dy == 0 && dx == 0) continue;
          const int x2 = x + dx;
          if (x2 < 0 || x2 >= W_) continue;
          if (sh[ls + dy][x2] > v) want = false;
        }
      }
    }

    const unsigned long long mball = __ballot(want);
    if (mball == 0ull) continue;
    if (want) {
      const float score = 1.0f / (1.0f + __expf(-v));
      const uint32_t idx = (uint32_t)c * (H_ * W_) + (uint32_t)r * W_ + (uint32_t)x;
      const unsigned long long key =
          ((unsigned long long)__float_as_uint(score) << 32) |
          (unsigned long long)(~idx);

      const unsigned lane = threadIdx.x & 31u;
      const int leader = __ffsll(mball) - 1;
      unsigned base = 0u;
      if ((int)lane == leader)
        base = atomicAdd(&cnt[b], (unsigned)__popcll(mball));   // 1 atomic / wave
      base = __shfl(base, leader, 32);
      const unsigned pos = base + (unsigned)__popcll(mball & ((1ull << lane) - 1ull));
      if (pos < CAND_CAP) cand[(size_t)b * CAND_CAP + pos] = key;
    }
  }
}

// ---------------------------------------------------------------------------
// One 1024-thread workgroup per batch: 100 iterated max-reductions over the
// candidate list (L2-resident). Wave32 shfl reduction + 32-entry LDS stage.
// Keys are unique -> deterministic result.
// ---------------------------------------------------------------------------
__global__ __launch_bounds__(1024)
void topk_kernel(unsigned long long* __restrict__ cand,
                 const unsigned int* __restrict__ cnt,
                 unsigned long long* __restrict__ selected) {
  const int b = blockIdx.x;
  unsigned int n = cnt[b];
  if (n > CAND_CAP) n = CAND_CAP;
  unsigned long long* cb = cand + (size_t)b * CAND_CAP;

  __shared__ unsigned long long warpred[32];
  const int tid  = threadIdx.x;
  const int lane = tid & 31;
  const int wid  = tid >> 5;

  for (int k = 0; k < K_; ++k) {
    unsigned long long m = 0ull;
    for (unsigned int i = (unsigned)tid; i < n; i += 1024u) {
      const unsigned long long v = cb[i];
      if (v > m) m = v;
    }
    #pragma unroll
    for (int off = 16; off > 0; off >>= 1) {
      const unsigned long long t = __shfl_down(m, off, 32);
      if (t > m) m = t;
    }
    if (lane == 0) warpred[wid] = m;
    __syncthreads();
    if (wid == 0) {
      unsigned long long v = warpred[lane];
      #pragma unroll
      for (int off = 16; off > 0; off >>= 1) {
        const unsigned long long t = __shfl_down(v, off, 32);
        if (t > v) v = t;
      }
      if (lane == 0) warpred[0] = v;
    }
    __syncthreads();
    const unsigned long long mx = warpred[0];
    if (tid == 0) selected[(size_t)b * K_ + k] = mx;
    if (mx != 0ull) {                          // remove winner (unique key)
      for (unsigned int i = (unsigned)tid; i < n; i += 1024u) {
        if (cb[i] == mx) cb[i] = 0ull;
      }
    }
    __syncthreads();
  }
}

// ---------------------------------------------------------------------------
// One thread per detection slot: sparse gather of 50 channels + full decode.
// Avoids the reference's 98 MB take_along_axis materialization (only 320 KB
// of pred_reg is ever touched; all of it L2-resident).
// ---------------------------------------------------------------------------
__global__ __launch_bounds__(256)
void decode_kernel(const float* __restrict__ reg,
                   const float* __restrict__ calib,
                   const float* __restrict__ pad_size,
                   const float* __restrict__ dim_mean,
                   const unsigned long long* __restrict__ selected,
                   float* __restrict__ out) {
  const int t = blockIdx.x * blockDim.x + threadIdx.x;
  if (t >= B_ * K_) return;
  const int b = t / K_;

  const unsigned long long key = selected[t];
  const float score = __uint_as_float((uint32_t)(key >> 32));
  float* o = out + (size_t)t * NCOL_;
  if (key == 0ull || !(score >= 0.2f)) {       // masked row -> zeros
    #pragma unroll
    for (int i = 0; i < NCOL_; ++i) o[i] = 0.0f;
    return;
  }

  const uint32_t idx = ~(uint32_t)(key & 0xFFFFFFFFu);
  const int c   = (int)(idx / (H_ * W_));
  const int rem = (int)(idx - (uint32_t)c * (H_ * W_));
  const int y   = rem / W_;
  const int x   = rem - y * W_;

  float pois[CREG_];
  const float* rp = reg + (size_t)b * CREG_ * H_ * W_ + rem;
  #pragma unroll
  for (int ch = 0; ch < CREG_; ++ch) pois[ch] = rp[(size_t)ch * (H_ * W_)];

  const float fu = calib[b * 4 + 0], cu = calib[b * 4 + 1];
  const float fv = calib[b * 4 + 2], cv = calib[b * 4 + 3];
  const float pad0 = pad_size[b * 2 + 0], pad1 = pad_size[b * 2 + 1];
  const float xs = (float)x, ys = (float)y;

  const float r2d0 = fmaxf(pois[0], 0.f), r2d1 = fmaxf(pois[1], 0.f);
  const float r2d2 = fmaxf(pois[2], 0.f), r2d3 = fmaxf(pois[3], 0.f);
  const float off0 = pois[4], off1 = pois[5];
  const float dim0 = __expf(pois[6]) * dim_mean[c * 3 + 0];
  const float dim1 = __expf(pois[7]) * dim_mean[c * 3 + 1];
  const float dim2 = __expf(pois[8]) * dim_mean[c * 3 + 2];
  const float h3d  = dim1;

  // keypoint y coords: kpt[j].y = pois[26 + 2j]
  const float ky0 = pois[26], ky1 = pois[28], ky2 = pois[30], ky3 = pois[32];
  const float ky4 = pois[34], ky5 = pois[36], ky6 = pois[38], ky7 = pois[40];
  const float ky8 = pois[42], ky9 = pois[44];
  const float cu0 = __expf(pois[45]), cu1 = __expf(pois[46]), cu2 = __expf(pois[47]);
  const float du  = __expf(pois[49]);

  const float sig = 1.0f / (1.0f + __expf(-pois[48]));
  const float ddirect = fminf(fmaxf(1.0f / sig - 1.0f, 0.1f), 100.0f);

  const float fh = fu * h3d;
  float d_ctr = fh / (fmaxf(ky8 - ky9, 0.f) * (float)DOWN_ + 1e-6f);
  float d_02  = 0.5f * (fh / (fmaxf(ky0 - ky4, 0.f) * (float)DOWN_ + 1e-6f) +
                        fh / (fmaxf(ky2 - ky6, 0.f) * (float)DOWN_ + 1e-6f));
  float d_13  = 0.5f * (fh / (fmaxf(ky1 - ky5, 0.f) * (float)DOWN_ + 1e-6f) +
                        fh / (fmaxf(ky3 - ky7, 0.f) * (float)DOWN_ + 1e-6f));
  d_ctr = fminf(fmaxf(d_ctr, 0.1f), 100.0f);
  d_02  = fminf(fmaxf(d_02,  0.1f), 100.0f);
  d_13  = fminf(fmaxf(d_13,  0.1f), 100.0f);

  const float w0 = 1.0f / du, w1 = 1.0f / cu0, w2 = 1.0f / cu1, w3 = 1.0f / cu2;
  const float wsum = w0 + w1 + w2 + w3;
  const float depth = (ddirect * w0 + d_ctr * w1 + d_02 * w2 + d_13 * w3) / wsum;
  const float depth_err = (w0 * du + w1 * cu0 + w2 * cu1 + w3 * cu2) / wsum;

  const float bx0 = fminf(fmaxf((xs - r2d0) * (float)DOWN_ - pad0, 0.f), (float)(IMG_W_ - 1));
  const float by0 = fminf(fmaxf((ys - r2d1) * (float)DOWN_ - pad1, 0.f), (float)(IMG_H_ - 1));
  const float bx1 = fminf(fmaxf((xs + r2d2) * (float)DOWN_ - pad0, 0.f), (float)(IMG_W_ - 1));
  const float by1 = fminf(fmaxf((ys + r2d3) * (float)DOWN_ - pad1, 0.f), (float)(IMG_H_ - 1));

  const float px = (xs + off0) * (float)DOWN_ - pad0;
  const float py = (ys + off1) * (float)DOWN_ - pad1;
  const float x3 = (px - cu) * depth / fu;
  const float y3 = (py - cv) * depth / fv;

  o[0]  = (float)c;   // cls
  o[1]  = score;      // score
  o[2]  = bx0;  o[3] = by0;  o[4] = bx1;  o[5] = by1;   // box2d
  o[6]  = dim0; o[7] = dim1; o[8] = dim2;               // dims
  o[9]  = x3;   o[10] = y3;  o[11] = depth;             // loc3d
  o[12] = depth_err;                                    // depth_err
}

// ---------------------------------------------------------------------------
extern "C" void kernel_launch(void* const* d_in, const int* in_sizes, int n_in,
                              void* d_out, int out_size, void* d_ws, size_t ws_size,
                              hipStream_t stream) {
  (void)in_sizes; (void)n_in; (void)out_size; (void)ws_size;
  const float* hmp   = (const float*)d_in[0];
  const float* reg   = (const float*)d_in[1];
  const float* calib = (const float*)d_in[2];
  const float* padsz = (const float*)d_in[3];
  const float* dmean = (const float*)d_in[4];
  float* out = (float*)d_out;

  uint8_t* ws = (uint8_t*)d_ws;
  unsigned int* cnt            = (unsigned int*)ws;                       // 16 u32
  unsigned long long* selected = (unsigned long long*)(ws + 256);         // 1600 u64
  unsigned long long* cand     = (unsigned long long*)(ws + 16384);       // 16*CAND_CAP u64

  hipLaunchKernelGGL(init_kernel, dim3(1), dim3(32), 0, stream, cnt);
  hipLaunchKernelGGL(nms_topcand_kernel,
                     dim3(B_ * NCLS_ * (H_ / ROWS_PER_BLK)), dim3(256), 0, stream,
                     hmp, cand, cnt);
  hipLaunchKernelGGL(topk_kernel, dim3(B_), dim3(1024), 0, stream, cand, cnt, selected);
  hipLaunchKernelGGL(decode_kernel, dim3((B_ * K_ + 255) / 256), dim3(256), 0, stream,
                     reg, calib, padsz, dmean, selected, out);
}